// YOLOLoss_25211458028176
// MI455X (gfx1250) — compile-verified
//
#include <hip/hip_runtime.h>

// YOLOv1 loss, forward only.
// preds:  [B, 7, 7, 30] f32   labels: [B, 7, 7, 30] f32   out: scalar f32
// Memory-bound: ~193 MB single-pass stream -> ~8.3 us floor at 23.3 TB/s.

typedef __attribute__((ext_vector_type(2))) float v2f;
typedef __attribute__((ext_vector_type(8))) float v8f;

__device__ __forceinline__ float sq(float x) { return x * x; }

// Matches reference _iou exactly (center-format boxes, no epsilon).
__device__ __forceinline__ float iou_ref(float acx, float acy, float aw, float ah,
                                         float bcx, float bcy, float bw, float bh)
{
    float L  = fmaxf(acx - 0.5f * aw, bcx - 0.5f * bw);
    float R  = fminf(acx + 0.5f * aw, bcx + 0.5f * bw);
    float T  = fmaxf(acy - 0.5f * ah, bcy - 0.5f * bh);
    float Bo = fminf(acy + 0.5f * ah, bcy + 0.5f * bh);
    float inter = fmaxf(R - L, 0.0f) * fmaxf(Bo - T, 0.0f);
    float uni   = aw * ah + bw * bh - inter;
    return inter / uni;
}

// Deterministic 256-thread block sum.
// Wave-level sum via V_WMMA_F32_16X16X4_F32 with an all-ones B matrix:
//   A (16x4 f32): VGPR0 = lane value (lanes 0-15 -> K=0, lanes 16-31 -> K=2),
//                 VGPR1 = 0. With B == ones, D[m][n] = rowsum[m] for all n.
//   C/D layout: lane in [0,15]: d[v] = rowsum[v]; lane in [16,31]: d[v] = rowsum[v+8].
//   Sum d[0..7] then xor-shuffle across the half-waves -> full wave sum.
// Requires EXEC all ones: called unconditionally by all 256 threads.
__device__ __forceinline__ float block_sum_256(float v, float* s8)
{
    v2f a; a[0] = v;    a[1] = 0.0f;
    v2f b; b[0] = 1.0f; b[1] = 1.0f;
    v8f c = {};
    v8f d = __builtin_amdgcn_wmma_f32_16x16x4_f32(
        /*neg_a=*/false, a, /*neg_b=*/false, b,
        /*c_mod=*/(short)0, c, /*reuse_a=*/false, /*reuse_b=*/false);
    float r = ((d[0] + d[1]) + (d[2] + d[3])) + ((d[4] + d[5]) + (d[6] + d[7]));
    r += __shfl_xor(r, 16, 32);            // combine the two half-wave partials
    if ((threadIdx.x & 31u) == 0u) s8[threadIdx.x >> 5] = r;
    __syncthreads();
    float t = 0.0f;
    if (threadIdx.x == 0) {
#pragma unroll
        for (int i = 0; i < 8; ++i) t += s8[i];   // fixed order: deterministic
    }
    return t;  // valid on thread 0 only
}

__global__ void __launch_bounds__(256)
yolo_cell_kernel(const float* __restrict__ preds,
                 const float* __restrict__ labels,
                 float* __restrict__ partial,
                 int n_cells)
{
    __shared__ float s8[8];
    float acc = 0.0f;
    const int stride = (int)(gridDim.x * blockDim.x);

    for (int cell = (int)(blockIdx.x * blockDim.x + threadIdx.x);
         cell < n_cells; cell += stride)
    {
        const v2f* pp = (const v2f*)(preds  + (size_t)cell * 30u);  // 8B aligned: 120 % 8 == 0
        const v2f* lp = (const v2f*)(labels + (size_t)cell * 30u);

        // Box + confidence channels [0..9]: 5 x b64 per array, non-temporal stream.
        v2f pv[5], lv[5];
#pragma unroll
        for (int i = 0; i < 5; ++i) {
            pv[i] = __builtin_nontemporal_load(pp + i);
            lv[i] = __builtin_nontemporal_load(lp + i);
        }
        // Class channels [10..29]: accumulate on the fly (keeps VGPR pressure low).
        float cls = 0.0f;
#pragma unroll
        for (int i = 5; i < 15; ++i) {
            v2f a = __builtin_nontemporal_load(pp + i);
            v2f b = __builtin_nontemporal_load(lp + i);
            float d0 = a[0] - b[0], d1 = a[1] - b[1];
            cls += d0 * d0 + d1 * d1;
        }

        float p0 = pv[0][0], p1 = pv[0][1], p2 = pv[1][0], p3 = pv[1][1], p4 = pv[2][0];
        float p5 = pv[2][1], p6 = pv[3][0], p7 = pv[3][1], p8 = pv[4][0], p9 = pv[4][1];
        float l0 = lv[0][0], l1 = lv[0][1], l2 = lv[1][0], l3 = lv[1][1], l4 = lv[2][0];
        float l5 = lv[2][1], l6 = lv[3][0], l7 = lv[3][1], l8 = lv[4][0];

        // grid offsets: x indexes dim 1, y indexes dim 2 (cell = (b*7 + x)*7 + y)
        int rem = cell % 49;
        float gx = (float)(rem / 7);
        float gy = (float)(rem % 7);

        float bcx = (l0 + gx) / 7.0f;
        float bcy = (l1 + gy) / 7.0f;
        float iou1 = iou_ref((p0 + gx) / 7.0f, (p1 + gy) / 7.0f, p2, p3, bcx, bcy, l2, l3);
        float iou2 = iou_ref((p5 + gx) / 7.0f, (p6 + gy) / 7.0f, p7, p8, bcx, bcy, l2, l3);

        bool w = iou1 > iou2;  // tie -> box 2, as in reference

        float xy1 = sq(p0 - l0) + sq(p1 - l1);
        float xy2 = sq(p5 - l5) + sq(p6 - l6);           // labels[...,5:7]!
        float wh1 = sq(sqrtf(p2) - sqrtf(l2)) + sq(sqrtf(p3) - sqrtf(l3));
        float wh2 = sq(sqrtf(p7) - sqrtf(l7)) + sq(sqrtf(p8) - sqrtf(l8));
        float o1  = sq(p4 - iou1);
        float o2  = sq(p9 - iou2);
        // obj + noobj cross terms: w ? (o1 + 0.5*o2) : (o2 + 0.5*o1)
        float objterm = w ? (o1 + 0.5f * o2) : (o2 + 0.5f * o1);

        float per_obj   = 5.0f * (w ? xy1 : xy2) + (w ? wh1 : wh2) + objterm + cls;
        float per_noobj = 0.5f * (p4 * p4 + p9 * p9);

        acc += (l4 == 1.0f) ? per_obj : per_noobj;
    }

    float t = block_sum_256(acc, s8);
    if (threadIdx.x == 0) partial[blockIdx.x] = t;
}

__global__ void __launch_bounds__(256)
yolo_final_kernel(const float* __restrict__ partial, int n,
                  float inv_B, float* __restrict__ out)
{
    __shared__ float s8[8];
    float acc = 0.0f;
    for (int i = (int)threadIdx.x; i < n; i += 256) acc += partial[i];
    float t = block_sum_256(acc, s8);
    if (threadIdx.x == 0) out[0] = t * inv_B;  // B = 16384 (pow2) -> exact
}

extern "C" void kernel_launch(void* const* d_in, const int* in_sizes, int n_in,
                              void* d_out, int out_size, void* d_ws, size_t ws_size,
                              hipStream_t stream)
{
    const float* preds  = (const float*)d_in[0];
    const float* labels = (const float*)d_in[1];
    float*       out    = (float*)d_out;
    float*       part   = (float*)d_ws;

    int n_cells = in_sizes[0] / 30;       // B*S*S
    int Bsz     = n_cells / 49;           // batch size

    int want = (n_cells + 255) / 256;
    int cap  = (int)(ws_size / sizeof(float));
    int nblocks = want < cap ? want : cap;
    if (nblocks < 1) nblocks = 1;

    yolo_cell_kernel<<<dim3(nblocks), dim3(256), 0, stream>>>(preds, labels, part, n_cells);
    yolo_final_kernel<<<dim3(1), dim3(256), 0, stream>>>(part, nblocks,
                                                         1.0f / (float)Bsz, out);
}